// RandomCrop3D_24824910971226
// MI455X (gfx1250) — compile-verified
//
#include <hip/hip_runtime.h>

// RandomCrop3D + trilinear resize, MI455X (gfx1250, wave32).
// Memory-bound: 268 MB streamed output vs 32 MB L2-resident input.
// Strategy: async global->LDS row staging (ASYNCcnt), double-buffered,
// trilinear interp from LDS, non-temporal output stores.

#define IMG  128
#define OUTD 64

typedef __attribute__((address_space(1))) int as1_int;
typedef __attribute__((address_space(3))) int as3_int;

#if __has_builtin(__builtin_amdgcn_global_load_async_to_lds_b32)
#define ASYNC_LDS 1
#else
#define ASYNC_LDS 0
#endif

__device__ __forceinline__ void async_copy_f32(const float* g, float* l) {
#if ASYNC_LDS
  // global_load_async_to_lds_b32: per-lane 4B DMA, tracked by ASYNCcnt
  __builtin_amdgcn_global_load_async_to_lds_b32((as1_int*)g, (as3_int*)l,
                                                /*imm offset*/0, /*cpol*/0);
#else
  *l = *g;  // synchronous fallback
#endif
}

__device__ __forceinline__ void wait_async_0() {
#if ASYNC_LDS
#if __has_builtin(__builtin_amdgcn_s_wait_asynccnt)
  __builtin_amdgcn_s_wait_asynccnt(0);
#else
  asm volatile("s_wait_asynccnt 0" ::: "memory");
#endif
#endif
}

__device__ __forceinline__ void wait_async_8() {
#if ASYNC_LDS
#if __has_builtin(__builtin_amdgcn_s_wait_asynccnt)
  __builtin_amdgcn_s_wait_asynccnt(8);
#else
  asm volatile("s_wait_asynccnt 8" ::: "memory");
#endif
#endif
}

// src = (j+0.5)*(sz/64) - 0.5, clipped to [0, sz-1]; i0=floor, i1=min(i0+1,sz-1)
__device__ __forceinline__ void axis_coord(int j, int sz, int& i0, int& i1,
                                           float& frac) {
  const float fsz = (float)sz;
  float src = ((float)j + 0.5f) * (fsz * (1.0f / 64.0f)) - 0.5f;
  src = fminf(fmaxf(src, 0.0f), fsz - 1.0f);
  const float fl = floorf(src);
  i0 = (int)fl;
  i1 = min(i0 + 1, sz - 1);
  frac = src - fl;
}

// Stage 16 rows (4 output-h values x {d0,d1} x {h0,h1}) for one h-chunk.
// Wave `wid` stages rows 2*wid and 2*wid+1; each row = 4 x 32-lane b32 DMAs.
__device__ __forceinline__ void stage_chunk(const float* __restrict__ vol,
                                            float (&buf)[16][IMG],
                                            int hb, int wid, int lane,
                                            int sz, int st_h, int st_w,
                                            int dg0, int dg1) {
#pragma unroll
  for (int rr = 0; rr < 2; ++rr) {
    const int r   = wid * 2 + rr;          // 0..15
    const int hh  = hb + (r >> 2);         // output h this row serves
    const int sel = r & 3;                 // 0:d0h0 1:d0h1 2:d1h0 3:d1h1
    int h0, h1; float wtmp;
    axis_coord(hh, sz, h0, h1, wtmp);
    const int hg = st_h + ((sel & 1) ? h1 : h0);
    const int dg = (sel & 2) ? dg1 : dg0;
    const float* grow = vol + ((size_t)dg * IMG + (size_t)hg) * IMG + st_w;
    float* lrow = buf[r];
#pragma unroll
    for (int sub = 0; sub < 4; ++sub) {
      const int k    = sub * 32 + lane;          // LDS slot 0..127
      const int ksrc = (k < sz) ? k : (sz - 1);  // clamp: stays inside row
      async_copy_f32(grow + ksrc, lrow + k);
    }
  }
}

__global__ void __launch_bounds__(256)
crop_resize_kernel(const float* __restrict__ x,
                   const int* __restrict__ sizes,
                   const int* __restrict__ starts,
                   float* __restrict__ out) {
  __shared__ float rows[2][16][IMG];  // 16 KB, double-buffered

  const int tid  = threadIdx.x;
  const int w    = tid & 63;
  const int hsub = (tid >> 6) & 3;   // which of 4 h rows in a chunk
  const int wid  = tid >> 5;         // wave id 0..7 (wave32)
  const int lane = tid & 31;

  const int blk  = blockIdx.x;
  const int d    = blk & (OUTD - 1);
  const int crop = blk >> 6;
  const int bvol = crop >> 6;        // 64 crops per batch volume

  const int sz   = sizes[crop];
  const int st_d = starts[crop * 3 + 0];
  const int st_h = starts[crop * 3 + 1];
  const int st_w = starts[crop * 3 + 2];

  const float* vol = x + (size_t)bvol * (IMG * IMG * IMG);

  // d-axis coords: uniform over the block
  int d0, d1; float wd;
  axis_coord(d, sz, d0, d1, wd);
  const int dg0 = st_d + d0;
  const int dg1 = st_d + d1;

  // w-axis coords: loop-invariant per thread
  int k0, k1; float ww;
  axis_coord(w, sz, k0, k1, ww);

  float* outp = out + (size_t)crop * (OUTD * OUTD * OUTD)
                    + (size_t)d * (OUTD * OUTD);

  // Pipeline prologue: stage chunk 0
  stage_chunk(vol, rows[0], 0, wid, lane, sz, st_h, st_w, dg0, dg1);

  for (int c = 0; c < 16; ++c) {
    const int cur = c & 1;
    if (c + 1 < 16) {
      // Issue next chunk into the other buffer, then wait so that only the
      // newest 8 DMAs may be outstanding -> chunk c is complete (in-order).
      stage_chunk(vol, rows[cur ^ 1], (c + 1) * 4, wid, lane,
                  sz, st_h, st_w, dg0, dg1);
      wait_async_8();
    } else {
      wait_async_0();
    }
    __syncthreads();

    const int h = c * 4 + hsub;
    int h0i, h1i; float wh;
    axis_coord(h, sz, h0i, h1i, wh);

    const float* r00 = rows[cur][hsub * 4 + 0];
    const float* r01 = rows[cur][hsub * 4 + 1];
    const float* r10 = rows[cur][hsub * 4 + 2];
    const float* r11 = rows[cur][hsub * 4 + 3];

    const float c00 = r00[k0] * (1.0f - ww) + r00[k1] * ww;
    const float c01 = r01[k0] * (1.0f - ww) + r01[k1] * ww;
    const float c10 = r10[k0] * (1.0f - ww) + r10[k1] * ww;
    const float c11 = r11[k0] * (1.0f - ww) + r11[k1] * ww;
    const float c0  = c00 * (1.0f - wh) + c01 * wh;
    const float c1  = c10 * (1.0f - wh) + c11 * wh;
    const float v   = c0 * (1.0f - wd) + c1 * wd;

    // Streamed 268 MB output: non-temporal, don't pollute L2
    __builtin_nontemporal_store(v, outp + (size_t)h * OUTD + w);

    __syncthreads();  // buffer `cur` gets overwritten next iteration+1
  }
}

extern "C" void kernel_launch(void* const* d_in, const int* in_sizes, int n_in,
                              void* d_out, int out_size, void* d_ws,
                              size_t ws_size, hipStream_t stream) {
  const float* x      = (const float*)d_in[0];
  const int*   sizes  = (const int*)d_in[1];
  const int*   starts = (const int*)d_in[2];
  float*       out    = (float*)d_out;

  const int n_crops = in_sizes[1];          // B*N = 256
  dim3 grid((unsigned)(n_crops * OUTD));    // one block per (crop, d-slice)
  crop_resize_kernel<<<grid, 256, 0, stream>>>(x, sizes, starts, out);
}